// CustomyoloLoss_8074538517107
// MI455X (gfx1250) — compile-verified
//
#include <hip/hip_runtime.h>
#include <hip/hip_bf16.h>

// YOLO loss: memory-bound streaming reduction.
// 176.6 MB read @ 23.3 TB/s -> ~7.6us roofline. Compute is negligible.
// Wave-level reduction is done on the (otherwise idle) matrix unit via
// V_WMMA_F32_16X16X4_F32 with B = ones: D[m][n] = s_m + s_{m+16} (exact f32).

typedef float v2f __attribute__((ext_vector_type(2)));
typedef float v8f __attribute__((ext_vector_type(8)));

__device__ __forceinline__ float wave_reduce_wmma(float s) {
    // A (16x4 f32, 2 VGPRs): lane l -> A[l%16][l<16 ? 0 : 2] = s_l ; K=1,3 rows zero.
    v2f a; a.x = s;    a.y = 0.0f;
    // B (4x16 f32) = all ones.
    v2f b; b.x = 1.0f; b.y = 1.0f;
    v8f c = {};
    // D[m][n] = sum_k A[m][k]*B[k][n] = s_m + s_{m+16}
    v8f d = __builtin_amdgcn_wmma_f32_16x16x4_f32(
        /*neg_a=*/false, a, /*neg_b=*/false, b,
        /*c_mod=*/(short)0, c, /*reuse_a=*/false, /*reuse_b=*/false);
    // C/D layout: vgpr j, lanes 0-15 -> row M=j ; lanes 16-31 -> row M=j+8.
    float partial = d[0] + d[1] + d[2] + d[3] + d[4] + d[5] + d[6] + d[7];
    // lanes 0-15 hold sum over M=0..7, lanes 16-31 hold sum over M=8..15.
    partial += __shfl_xor(partial, 16, 32);
    return partial; // full 32-lane sum, broadcast to all lanes (exact f32)
}

__device__ __forceinline__ float cell_loss(const float* __restrict__ p,
                                           const float* __restrict__ t) {
    // predictions per cell are 8B-aligned (cell stride 120B, row stride 5880B)
    float pv[30];
    const float2* p2 = reinterpret_cast<const float2*>(p);
#pragma unroll
    for (int k = 0; k < 15; ++k) {
        float2 v = p2[k];
        pv[2 * k]     = v.x;
        pv[2 * k + 1] = v.y;
    }
    float tv[25];
#pragma unroll
    for (int k = 0; k < 25; ++k) tv[k] = t[k];

    float c1 = pv[4], c2 = pv[9], cc = tv[4];
    bool present = (cc == 1.0f);
    bool resp1   = (c1 > c2);

    float dsel        = (resp1 ? c1 : c2) - cc;
    float obj_present = dsel * dsel;
    float obj_absent  = 0.5f * fmaf(c1, c1, c2 * c2);
    float obj         = present ? obj_present : obj_absent;

    float cls = 0.0f;
#pragma unroll
    for (int k = 0; k < 20; ++k) {
        float d = pv[10 + k] - tv[5 + k];
        cls = fmaf(d, d, cls);
    }

    float pc0 = resp1 ? pv[0] : pv[5];
    float pc1 = resp1 ? pv[1] : pv[6];
    float ph0 = resp1 ? pv[2] : pv[7];
    float ph1 = resp1 ? pv[3] : pv[8];
    float d0 = pc0 - tv[0];
    float d1 = pc1 - tv[1];
    float d2 = sqrtf(ph0) - sqrtf(tv[2]);
    float d3 = sqrtf(ph1) - sqrtf(tv[3]);
    float box = fmaf(d0, d0, fmaf(d1, d1, fmaf(d2, d2, d3 * d3)));

    return obj + (present ? fmaf(5.0f, box, cls) : 0.0f);
}

__global__ __launch_bounds__(256)
void yolo_loss_partials(const float* __restrict__ pred,
                        const float* __restrict__ targ,
                        float* __restrict__ block_partials,
                        int n_cells) {
    int cell = blockIdx.x * 256 + threadIdx.x;
    float s = 0.0f;
    if (cell < n_cells) {
        int b  = cell / 49;
        int ij = cell - b * 49;
        s = cell_loss(pred + (size_t)b * 1470 + ij * 30,
                      targ + (size_t)b * 1225 + ij * 25);
    }
    // EXEC is all-ones here (branch above has re-converged).
    float wsum = wave_reduce_wmma(s);

    __shared__ float lds[8];
    int wave = threadIdx.x >> 5;
    int lane = threadIdx.x & 31;
    if (lane == 0) lds[wave] = wsum;
    __syncthreads();
    if (threadIdx.x == 0) {
        float tot = 0.0f;
#pragma unroll
        for (int i = 0; i < 8; ++i) tot += lds[i];  // fixed order: deterministic
        block_partials[blockIdx.x] = tot;
    }
}

__global__ __launch_bounds__(256)
void yolo_loss_final(const float* __restrict__ block_partials,
                     float* __restrict__ out,
                     int n) {
    float s = 0.0f;
    for (int i = threadIdx.x; i < n; i += 256) s += block_partials[i];  // fixed order
    float wsum = wave_reduce_wmma(s);

    __shared__ float lds[8];
    int wave = threadIdx.x >> 5;
    int lane = threadIdx.x & 31;
    if (lane == 0) lds[wave] = wsum;
    __syncthreads();
    if (threadIdx.x == 0) {
        float tot = 0.0f;
#pragma unroll
        for (int i = 0; i < 8; ++i) tot += lds[i];
        out[0] = tot;
    }
}

extern "C" void kernel_launch(void* const* d_in, const int* in_sizes, int n_in,
                              void* d_out, int out_size, void* d_ws, size_t ws_size,
                              hipStream_t stream) {
    const float* pred = (const float*)d_in[0];
    const float* targ = (const float*)d_in[1];
    float* out = (float*)d_out;

    int B = in_sizes[0] / 1470;        // 16384
    int n_cells = B * 49;              // 802816
    int nblk = (n_cells + 255) / 256;  // 3136

    float* partials = (float*)d_ws;    // nblk * 4 bytes = 12.5 KB, well under ws_size

    yolo_loss_partials<<<nblk, 256, 0, stream>>>(pred, targ, partials, n_cells);
    yolo_loss_final<<<1, 256, 0, stream>>>(partials, out, nblk);
}